// MultiTargetSearchGNN_81372450390530
// MI455X (gfx1250) — compile-verified
//
#include <hip/hip_runtime.h>
#include <hip/hip_bf16.h>
#include <math.h>

#define N_NODES 50000
#define F_H     128
#define N_EDGES 800000
#define N_GROUPS 64
#define GN_EPS  1e-5f
#define NH      ((size_t)N_NODES * F_H)
#define GH      (N_GROUPS * F_H)

typedef __attribute__((ext_vector_type(2))) float v2f;
typedef __attribute__((ext_vector_type(8))) float v8f;

// ---------------- GEMM: Y[M x 128] = X[M x 128] @ W[128 x 128] (+bias)(+relu) ---
// One block = 16 rows; 8 waves = 8 column tiles of 16. 32 k-steps of
// V_WMMA_F32_16X16X4_F32 per wave (exact fp32 accumulation).
template <bool BIAS, bool RELU>
__global__ __launch_bounds__(256) void gemm128_wmma(
    const float* __restrict__ X, const float* __restrict__ W,
    float* __restrict__ Y, const float* __restrict__ bias)
{
    const int wave = threadIdx.x >> 5;       // 0..7 -> column tile
    const int lane = threadIdx.x & 31;
    const int half = lane >> 4;              // K sub-pair selector
    const int l    = lane & 15;              // M (for A) / N (for B,C)
    const int rowBase = blockIdx.x << 4;
    const int colBase = wave << 4;
    const float* Xr = X + (size_t)rowBase * F_H;

    v8f c = {0.f, 0.f, 0.f, 0.f, 0.f, 0.f, 0.f, 0.f};
#pragma unroll 4
    for (int k0 = 0; k0 < F_H; k0 += 4) {
        const int ka = k0 + 2 * half;
        v2f a, b;
        // A 16x4: VGPR0 = {K=k0 (lanes0-15), K=k0+2 (lanes16-31)}, VGPR1 = K+1
        a.x = Xr[l * F_H + ka];
        a.y = Xr[l * F_H + ka + 1];
        // B 4x16: rows striped across lanes within a VGPR
        b.x = W[ka * F_H + colBase + l];
        b.y = W[(ka + 1) * F_H + colBase + l];
        c = __builtin_amdgcn_wmma_f32_16x16x4_f32(false, a, false, b,
                                                  (short)0, c, false, false);
    }
    float bcol = 0.0f;
    if (BIAS) bcol = bias[colBase + l];
#pragma unroll
    for (int r = 0; r < 8; ++r) {
        const int row = rowBase + r + 8 * half;   // C: VGPR r -> M=r / M=r+8
        const int col = colBase + l;
        float v = c[r];
        if (BIAS) v += bcol;
        if (RELU) v = fmaxf(v, 0.0f);
        Y[(size_t)row * F_H + col] = v;
    }
}

// ---------------- degree / normalization scalars ------------------------------
__global__ void edge_deg(const int* __restrict__ dst, float* __restrict__ deg)
{
    int e = blockIdx.x * blockDim.x + threadIdx.x;
    if (e < N_EDGES) atomicAdd(&deg[dst[e]], 1.0f);
}

__global__ void node_dis(const float* __restrict__ deg,
                         float* __restrict__ dis, float* __restrict__ dis2)
{
    int i = blockIdx.x * blockDim.x + threadIdx.x;
    if (i < N_NODES) {
        float d = deg[i] + 1.0f;          // self-loop
        dis[i]  = rsqrtf(d);
        dis2[i] = 1.0f / d;
    }
}

__global__ void grp_cnt(const int* __restrict__ batch, float* __restrict__ cnt)
{
    int i = blockIdx.x * blockDim.x + threadIdx.x;
    if (i < N_NODES) atomicAdd(&cnt[batch[i]], 1.0f);
}

// ---------------- edge scatter: AG[dst] += H[src] * dis[src]*dis[dst] ----------
__global__ __launch_bounds__(256) void edge_scatter(
    const float* __restrict__ H, const int* __restrict__ src,
    const int* __restrict__ dst, const float* __restrict__ dis,
    float* __restrict__ AG)
{
    const int lane = threadIdx.x & 31;
    const int wid  = (blockIdx.x * blockDim.x + threadIdx.x) >> 5;
    const int nwav = (gridDim.x * blockDim.x) >> 5;
    for (int e = wid; e < N_EDGES; e += nwav) {
        // e is wave-uniform: force indices scalar so addressing is SGPR-based
        const int s = __builtin_amdgcn_readfirstlane(src[e]);
        const int d = __builtin_amdgcn_readfirstlane(dst[e]);
        const float nrm = dis[s] * dis[d];
        const float4 v = ((const float4*)(H + (size_t)s * F_H))[lane];
        float* a = AG + (size_t)d * F_H + lane * 4;
        atomicAdd(a + 0, v.x * nrm);
        atomicAdd(a + 1, v.y * nrm);
        atomicAdd(a + 2, v.z * nrm);
        atomicAdd(a + 3, v.w * nrm);
    }
}

// ---------------- self-loop + bias: AG[i] += H[i]*dis2[i] + b -----------------
__global__ __launch_bounds__(256) void selfloop_bias(
    float* __restrict__ AG, const float* __restrict__ H,
    const float* __restrict__ dis2, const float* __restrict__ bias)
{
    const int idx = blockIdx.x * blockDim.x + threadIdx.x;
    const int wid = idx >> 5, lane = idx & 31;
    if (wid >= N_NODES) return;
    const float s = dis2[wid];
    const float4 h = ((const float4*)(H + (size_t)wid * F_H))[lane];
    const float4 b = ((const float4*)bias)[lane];
    float4* a = (float4*)(AG + (size_t)wid * F_H) + lane;
    float4 v = *a;
    v.x += h.x * s + b.x;  v.y += h.y * s + b.y;
    v.z += h.z * s + b.z;  v.w += h.w * s + b.w;
    *a = v;
}

// ---------------- GraphNorm: per-group sum / sumsq ----------------------------
// Block = 8 consecutive nodes (batch is sorted, so almost always one group):
// LDS pre-reduction (ds_add_f32) then one global-atomic flush per block.
__global__ __launch_bounds__(256) void gn_reduce(
    const float* __restrict__ Y, const int* __restrict__ batch,
    float* __restrict__ gsum, float* __restrict__ gsq)
{
    __shared__ float ls[F_H];
    __shared__ float lq[F_H];
    const int t = threadIdx.x;
    if (t < F_H) { ls[t] = 0.0f; lq[t] = 0.0f; }
    __syncthreads();

    const int wave = t >> 5, lane = t & 31;
    const int node = blockIdx.x * 8 + wave;          // 6250 * 8 == 50000 exact
    const int g0   = batch[blockIdx.x * 8];
    const int g    = batch[node];
    const float4 v = ((const float4*)(Y + (size_t)node * F_H))[lane];
    if (g == g0) {
        float* s = ls + lane * 4;
        float* q = lq + lane * 4;
        atomicAdd(s + 0, v.x); atomicAdd(q + 0, v.x * v.x);
        atomicAdd(s + 1, v.y); atomicAdd(q + 1, v.y * v.y);
        atomicAdd(s + 2, v.z); atomicAdd(q + 2, v.z * v.z);
        atomicAdd(s + 3, v.w); atomicAdd(q + 3, v.w * v.w);
    } else {  // rare group-boundary block: direct global atomics
        float* s = gsum + g * F_H + lane * 4;
        float* q = gsq  + g * F_H + lane * 4;
        atomicAdd(s + 0, v.x); atomicAdd(q + 0, v.x * v.x);
        atomicAdd(s + 1, v.y); atomicAdd(q + 1, v.y * v.y);
        atomicAdd(s + 2, v.z); atomicAdd(q + 2, v.z * v.z);
        atomicAdd(s + 3, v.w); atomicAdd(q + 3, v.w * v.w);
    }
    __syncthreads();
    if (t < F_H) {
        atomicAdd(&gsum[g0 * F_H + t], ls[t]);
        atomicAdd(&gsq [g0 * F_H + t], lq[t]);
    }
}

// var(x - a*m) = E[x^2] - (2a - a^2) * m^2   (exactly the reference's variance)
__global__ void gn_finalize(const float* __restrict__ gsum, const float* __restrict__ gsq,
                            const float* __restrict__ gcnt, const float* __restrict__ alpha,
                            float* __restrict__ gam_mean, float* __restrict__ ginvstd)
{
    const int g = blockIdx.x, f = threadIdx.x;
    const float cnt  = fmaxf(gcnt[g], 1.0f);
    const float mean = gsum[g * F_H + f] / cnt;
    const float ex2  = gsq [g * F_H + f] / cnt;
    const float a    = alpha[f];
    const float var  = ex2 - (2.0f * a - a * a) * mean * mean;
    gam_mean[g * F_H + f] = a * mean;
    ginvstd [g * F_H + f] = rsqrtf(fmaxf(var, 0.0f) + GN_EPS);
}

// out = [residual +] relu((y - a*mean) * invstd * gamma + beta)
__global__ __launch_bounds__(256) void gn_apply(
    const float* __restrict__ Y, const int* __restrict__ batch,
    const float* __restrict__ gam_mean, const float* __restrict__ ginvstd,
    const float* __restrict__ gamma, const float* __restrict__ beta,
    const float* __restrict__ residual, float* __restrict__ out)
{
    const int idx = blockIdx.x * blockDim.x + threadIdx.x;
    const int wid = idx >> 5, lane = idx & 31;
    if (wid >= N_NODES) return;
    const int g = batch[wid];
    const float4 v  = ((const float4*)(Y + (size_t)wid * F_H))[lane];
    const float4 m  = ((const float4*)(gam_mean + g * F_H))[lane];
    const float4 is = ((const float4*)(ginvstd + g * F_H))[lane];
    const float4 gm = ((const float4*)gamma)[lane];
    const float4 bt = ((const float4*)beta)[lane];
    float4 o;
    o.x = fmaxf((v.x - m.x) * is.x * gm.x + bt.x, 0.0f);
    o.y = fmaxf((v.y - m.y) * is.y * gm.y + bt.y, 0.0f);
    o.z = fmaxf((v.z - m.z) * is.z * gm.z + bt.z, 0.0f);
    o.w = fmaxf((v.w - m.w) * is.w * gm.w + bt.w, 0.0f);
    if (residual) {
        const float4 r = ((const float4*)(residual + (size_t)wid * F_H))[lane];
        o.x += r.x; o.y += r.y; o.z += r.z; o.w += r.w;
    }
    ((float4*)(out + (size_t)wid * F_H))[lane] = o;
}

// ---------------- head: out[i] = sigmoid(dot(T[i], Wp2) + bp2) ----------------
__global__ __launch_bounds__(256) void head_dot_sigmoid(
    const float* __restrict__ T, const float* __restrict__ Wp2,
    const float* __restrict__ bp2, float* __restrict__ out)
{
    const int idx = blockIdx.x * blockDim.x + threadIdx.x;
    const int wid = idx >> 5, lane = idx & 31;
    if (wid >= N_NODES) return;
    const float4 t = ((const float4*)(T + (size_t)wid * F_H))[lane];
    const float4 w = ((const float4*)Wp2)[lane];
    float s = t.x * w.x + t.y * w.y + t.z * w.z + t.w * w.w;
#pragma unroll
    for (int off = 16; off > 0; off >>= 1) s += __shfl_down(s, off, 32);
    if (lane == 0) out[wid] = 1.0f / (1.0f + expf(-(s + bp2[0])));
}

// ---------------- per-layer driver --------------------------------------------
static void run_layer(const float* Xin, const float* W, const float* b,
                      const float* alpha, const float* gamma, const float* beta,
                      const float* residual, float* Hbuf, float* Outbuf, float* AG,
                      const int* src, const int* dst,
                      const float* dis, const float* dis2, const int* batch,
                      float* gsum, float* gsq, const float* gcnt,
                      float* gam, float* giv, hipStream_t stream)
{
    const int nodeWaveBlocks = (N_NODES * 32 + 255) / 256;
    gemm128_wmma<false, false><<<N_NODES / 16, 256, 0, stream>>>(Xin, W, Hbuf, nullptr);
    hipMemsetAsync(AG, 0, NH * sizeof(float), stream);
    hipMemsetAsync(gsum, 0, 2 * GH * sizeof(float), stream);   // gsum+gsq contiguous
    edge_scatter<<<8192, 256, 0, stream>>>(Hbuf, src, dst, dis, AG);
    selfloop_bias<<<nodeWaveBlocks, 256, 0, stream>>>(AG, Hbuf, dis2, b);
    gn_reduce<<<N_NODES / 8, 256, 0, stream>>>(AG, batch, gsum, gsq);
    gn_finalize<<<N_GROUPS, F_H, 0, stream>>>(gsum, gsq, gcnt, alpha, gam, giv);
    gn_apply<<<nodeWaveBlocks, 256, 0, stream>>>(AG, batch, gam, giv, gamma, beta,
                                                 residual, Outbuf);
}

extern "C" void kernel_launch(void* const* d_in, const int* in_sizes, int n_in,
                              void* d_out, int out_size, void* d_ws, size_t ws_size,
                              hipStream_t stream)
{
    (void)in_sizes; (void)n_in; (void)out_size; (void)ws_size;
    const float* x    = (const float*)d_in[0];
    const int*   edge = (const int*)d_in[1];
    const int*   bat  = (const int*)d_in[2];
    const float* W1  = (const float*)d_in[3];
    const float* b1  = (const float*)d_in[4];
    const float* a1  = (const float*)d_in[5];
    const float* g1  = (const float*)d_in[6];
    const float* be1 = (const float*)d_in[7];
    const float* Wm  = (const float*)d_in[8];
    const float* bm  = (const float*)d_in[9];
    const float* am  = (const float*)d_in[10];
    const float* gm  = (const float*)d_in[11];
    const float* bem = (const float*)d_in[12];
    const float* WL  = (const float*)d_in[13];
    const float* bL  = (const float*)d_in[14];
    const float* aL  = (const float*)d_in[15];
    const float* gL  = (const float*)d_in[16];
    const float* beL = (const float*)d_in[17];
    const float* Wp1 = (const float*)d_in[18];
    const float* bp1 = (const float*)d_in[19];
    const float* Wp2 = (const float*)d_in[20];
    const float* bp2 = (const float*)d_in[21];
    const int* src = edge;
    const int* dst = edge + N_EDGES;

    // workspace carve (~78 MB)
    float* P0   = (float*)d_ws;
    float* P1   = P0 + NH;
    float* AG   = P1 + NH;
    float* deg  = AG + NH;
    float* dis  = deg + N_NODES;
    float* dis2 = dis + N_NODES;
    float* gsum = dis2 + N_NODES;
    float* gsq  = gsum + GH;
    float* gam  = gsq + GH;
    float* giv  = gam + GH;
    float* gcnt = giv + GH;

    const int nodeBlocks = (N_NODES + 255) / 256;
    const int edgeBlocks = (N_EDGES + 255) / 256;

    // degree -> dis / dis2 ; per-group node counts
    hipMemsetAsync(deg, 0, N_NODES * sizeof(float), stream);
    hipMemsetAsync(gcnt, 0, N_GROUPS * sizeof(float), stream);
    edge_deg<<<edgeBlocks, 256, 0, stream>>>(dst, deg);
    node_dis<<<nodeBlocks, 256, 0, stream>>>(deg, dis, dis2);
    grp_cnt<<<nodeBlocks, 256, 0, stream>>>(bat, gcnt);

    // layer 1 (no residual):   cur -> P0
    run_layer(x, W1, b1, a1, g1, be1, nullptr, P0, P0, AG,
              src, dst, dis, dis2, bat, gsum, gsq, gcnt, gam, giv, stream);
    // mid layer 0 (residual):  P0 -> P1
    run_layer(P0, Wm, bm, am, gm, bem, P0, P1, P1, AG,
              src, dst, dis, dis2, bat, gsum, gsq, gcnt, gam, giv, stream);
    // mid layer 1 (residual):  P1 -> P0
    run_layer(P1, Wm + F_H * F_H, bm + F_H, am + F_H, gm + F_H, bem + F_H,
              P1, P0, P0, AG,
              src, dst, dis, dis2, bat, gsum, gsq, gcnt, gam, giv, stream);
    // last layer (no residual): P0 -> P1
    run_layer(P0, WL, bL, aL, gL, beL, nullptr, P1, P1, AG,
              src, dst, dis, dis2, bat, gsum, gsq, gcnt, gam, giv, stream);

    // head: T = relu(P1 @ Wp1 + bp1) -> P0 ; out = sigmoid(T . Wp2 + bp2)
    gemm128_wmma<true, true><<<N_NODES / 16, 256, 0, stream>>>(P1, Wp1, P0, bp1);
    head_dot_sigmoid<<<(N_NODES * 32 + 255) / 256, 256, 0, stream>>>(
        P0, Wp2, bp2, (float*)d_out);
}